// MultiQueryMHA_2714419331130
// MI455X (gfx1250) — compile-verified
//
#include <hip/hip_runtime.h>
#include <math.h>

#define B_    1024
#define S_    200
#define H_    256
#define NH_   4
#define U_    5
#define D_    64

typedef __attribute__((ext_vector_type(2))) float v2f;
typedef __attribute__((ext_vector_type(4))) float v4f;
typedef __attribute__((ext_vector_type(8))) float v8f;

// ---------------------------------------------------------------------------
// Generic fp32 WMMA GEMM:  C[m, n] = sum_k A[m*lda + k] * W[n*K + k] + bias[n]
// A: M x K (row stride lda), W: N x K row-major, C: row stride ldc.
// Block = 128 threads = 4 waves; wave w owns 16 rows, grid.y picks a 128-wide
// N strip (8 accumulator tiles / wave). Uses V_WMMA_F32_16X16X4_F32.
//
// 8 k-values per pipeline step via one global_load_b128 per fragment:
//   half0 lanes hold k = kb..kb+3, half1 lanes hold k = kb+4..kb+7.
//   WMMA#0 uses (x,y)  -> k {kb,kb+1} (half0: slots K0/K1) + {kb+4,kb+5}
//   WMMA#1 uses (z,w)  -> k {kb+2,kb+3}               + {kb+6,kb+7}
// A and B use the identical k mapping, so the sum is exact (mod fp order).
// Double-buffered: next step's 9 b128 loads are issued before this step's
// 16 WMMAs, keeping loads in flight under the matrix pipe.
// ---------------------------------------------------------------------------
__global__ void __launch_bounds__(128) gemm_wmma_f32(
    const float* __restrict__ A, int lda,
    const float* __restrict__ W,
    const float* __restrict__ bias,
    float* __restrict__ C, int ldc,
    int K)
{
    const int wave = threadIdx.x >> 5;
    const int lane = threadIdx.x & 31;
    const int r    = lane & 15;
    const int half = lane >> 4;      // 0: lanes 0-15, 1: lanes 16-31
    const int m0   = (blockIdx.x * 4 + wave) * 16;
    const int n0   = blockIdx.y * 128;

    v8f acc[8];
#pragma unroll
    for (int t = 0; t < 8; ++t)
        acc[t] = (v8f){0.f,0.f,0.f,0.f,0.f,0.f,0.f,0.f};

    // per-lane base pointers, offset by half*4 (this half's 4-k chunk)
    const float* ap = A + (size_t)(m0 + r) * lda + half * 4;
    const float* wp = W + (size_t)(n0 + r) * K + half * 4;  // + t*16*K per tile

    // prologue: fetch k-step 0 (1 A + 8 B float4 fragments)
    v4f a_c, b_c[8];
    a_c = *(const v4f*)ap;
#pragma unroll
    for (int t = 0; t < 8; ++t)
        b_c[t] = *(const v4f*)(wp + (size_t)t * 16 * K);

    for (int kb = 0; kb + 8 < K; kb += 8) {
        const int kn = kb + 8;
        // issue next step's 9 b128 loads first (overlap with WMMAs below)
        v4f a_n, b_n[8];
        a_n = *(const v4f*)(ap + kn);
#pragma unroll
        for (int t = 0; t < 8; ++t)
            b_n[t] = *(const v4f*)(wp + (size_t)t * 16 * K + kn);

        // consume current fragments: 2 WMMAs per tile (lo = xy, hi = zw)
        v2f a_lo, a_hi;
        a_lo.x = a_c.x; a_lo.y = a_c.y;
        a_hi.x = a_c.z; a_hi.y = a_c.w;
#pragma unroll
        for (int t = 0; t < 8; ++t) {
            v2f b_lo; b_lo.x = b_c[t].x; b_lo.y = b_c[t].y;
            acc[t] = __builtin_amdgcn_wmma_f32_16x16x4_f32(
                false, a_lo, false, b_lo, (short)0, acc[t], false, false);
        }
#pragma unroll
        for (int t = 0; t < 8; ++t) {
            v2f b_hi; b_hi.x = b_c[t].z; b_hi.y = b_c[t].w;
            acc[t] = __builtin_amdgcn_wmma_f32_16x16x4_f32(
                false, a_hi, false, b_hi, (short)0, acc[t], false, false);
        }
        a_c = a_n;
#pragma unroll
        for (int t = 0; t < 8; ++t) b_c[t] = b_n[t];
    }
    // epilogue: last 8-k step
    {
        v2f a_lo, a_hi;
        a_lo.x = a_c.x; a_lo.y = a_c.y;
        a_hi.x = a_c.z; a_hi.y = a_c.w;
#pragma unroll
        for (int t = 0; t < 8; ++t) {
            v2f b_lo; b_lo.x = b_c[t].x; b_lo.y = b_c[t].y;
            acc[t] = __builtin_amdgcn_wmma_f32_16x16x4_f32(
                false, a_lo, false, b_lo, (short)0, acc[t], false, false);
        }
#pragma unroll
        for (int t = 0; t < 8; ++t) {
            v2f b_hi; b_hi.x = b_c[t].z; b_hi.y = b_c[t].w;
            acc[t] = __builtin_amdgcn_wmma_f32_16x16x4_f32(
                false, a_hi, false, b_hi, (short)0, acc[t], false, false);
        }
    }

    // C/D layout: VGPR j -> row m0 + j + half*8, col n0 + t*16 + r
#pragma unroll
    for (int t = 0; t < 8; ++t) {
        const int col = n0 + t * 16 + r;
        const float bv = bias ? bias[col] : 0.0f;
#pragma unroll
        for (int j = 0; j < 8; ++j) {
            const int row = m0 + j + half * 8;
            C[(size_t)row * ldc + col] = acc[t][j] + bv;
        }
    }
}

// ---------------------------------------------------------------------------
// Gather 5 history rows: cat[b, j*H + h] = input[b, idx_j, h]
// idx = clip(scaled_num - (j+1), -1, 200); idx<0 -> idx+S
// ---------------------------------------------------------------------------
__global__ void gather_kernel(const float* __restrict__ inp,
                              const int* __restrict__ scaled,
                              float* __restrict__ cat)
{
    const int b = blockIdx.x;
    const int h = threadIdx.x;
    const int sn = scaled[b];
#pragma unroll
    for (int j = 0; j < U_; ++j) {
        int idx = sn - (j + 1);
        if (idx > 200) idx = 200;
        if (idx < -1)  idx = -1;
        if (idx < 0)   idx += S_;
        cat[(size_t)b * (U_ * H_) + j * H_ + h] =
            inp[(size_t)b * (S_ * H_) + (size_t)idx * H_ + h];
    }
}

// ---------------------------------------------------------------------------
// mq /= (||mq||_2 over U axis + 1e-12), per (b, h)
// ---------------------------------------------------------------------------
__global__ void mqnorm_kernel(float* __restrict__ mq)
{
    const int b = blockIdx.x;
    const int h = threadIdx.x;
    float* p = mq + (size_t)b * (U_ * H_) + h;
    float v[U_];
    float s = 0.f;
#pragma unroll
    for (int u = 0; u < U_; ++u) { v[u] = p[u * H_]; s += v[u] * v[u]; }
    const float inv = 1.0f / (sqrtf(s) + 1e-12f);
#pragma unroll
    for (int u = 0; u < U_; ++u) p[u * H_] = v[u] * inv;
}

// ---------------------------------------------------------------------------
// Sigmoid attention logits with flat-reshape head semantics:
// L[b, s2, t=n*U+u] = sigmoid( sum_dd Q[b*1280 + n*320 + u*64 + dd]
//                                   * K[b*51200 + n*12800 + s2*64 + dd] )
// ---------------------------------------------------------------------------
__global__ void __launch_bounds__(256) logits_kernel(
    const float* __restrict__ Q, const float* __restrict__ Kb,
    float* __restrict__ L)
{
    __shared__ float qs[U_ * H_];  // 1280 floats
    const int b = blockIdx.x;
    for (int i = threadIdx.x; i < U_ * H_; i += 256)
        qs[i] = Q[(size_t)b * (U_ * H_) + i];
    __syncthreads();

    const float* kbase = Kb + (size_t)b * (S_ * H_);
    float* lbase = L + (size_t)b * (S_ * NH_ * U_);
    for (int i = threadIdx.x; i < S_ * NH_ * U_; i += 256) {
        const int s2 = i / (NH_ * U_);
        const int t  = i % (NH_ * U_);
        const int n  = t / U_;
        const int u  = t % U_;
        const float* kp = kbase + n * (S_ * D_) + s2 * D_;
        const float* qp = qs + n * (U_ * D_) + u * D_;
        float acc = 0.f;
#pragma unroll 8
        for (int dd = 0; dd < D_; ++dd) acc += qp[dd] * kp[dd];
        lbase[i] = 1.0f / (1.0f + expf(-acc));
    }
}

// ---------------------------------------------------------------------------
// Softmax over s2 of 2*L per (b,t); result written straight to att_map slice
// of d_out (same flat layout as am.reshape(B,NH,S,U)).
// ---------------------------------------------------------------------------
__global__ void softmax_seq_kernel(const float* __restrict__ L,
                                   float* __restrict__ SM)
{
    const int idx = blockIdx.x * blockDim.x + threadIdx.x; // (b, t)
    if (idx >= B_ * NH_ * U_) return;
    const int b = idx / (NH_ * U_);
    const int t = idx % (NH_ * U_);
    const float* col = L + (size_t)b * (S_ * NH_ * U_) + t;
    float mx = -1e30f;
    for (int s = 0; s < S_; ++s) mx = fmaxf(mx, 2.0f * col[s * (NH_ * U_)]);
    float sum = 0.f;
    for (int s = 0; s < S_; ++s) sum += expf(2.0f * col[s * (NH_ * U_)] - mx);
    const float inv = 1.0f / sum;
    float* o = SM + (size_t)b * (S_ * NH_ * U_) + t;
    for (int s = 0; s < S_; ++s)
        o[s * (NH_ * U_)] = expf(2.0f * col[s * (NH_ * U_)] - mx) * inv;
}

// ---------------------------------------------------------------------------
// L4-pool over U, mask -> -inf, softmax over S, NaN(all-masked) -> 0.
// One thread per (b, n).  am2: (B, S, NH)
// ---------------------------------------------------------------------------
__global__ void pooled_softmax_kernel(const float* __restrict__ SM,
                                      const int* __restrict__ mask,
                                      float* __restrict__ am2)
{
    const int idx = blockIdx.x * blockDim.x + threadIdx.x;
    if (idx >= B_ * NH_) return;
    const int b = idx / NH_;
    const int n = idx % NH_;
    const float* smb = SM + (size_t)b * (S_ * NH_ * U_) + n * U_;
    const int* mb = mask + (size_t)b * S_;

    float mx = -1e30f;
    bool any = false;
    for (int s = 0; s < S_; ++s) {
        if (!mb[s]) continue;
        const float* p = smb + s * (NH_ * U_);
        float s4 = 0.f;
#pragma unroll
        for (int u = 0; u < U_; ++u) { float v = p[u]; float v2 = v * v; s4 += v2 * v2; }
        const float z = 2.0f * sqrtf(sqrtf(s4));
        any = true;
        mx = fmaxf(mx, z);
    }
    float* ob = am2 + (size_t)b * (S_ * NH_) + n;
    if (!any) {
        for (int s = 0; s < S_; ++s) ob[s * NH_] = 0.0f;
        return;
    }
    float sum = 0.f;
    for (int s = 0; s < S_; ++s) {
        if (!mb[s]) continue;
        const float* p = smb + s * (NH_ * U_);
        float s4 = 0.f;
#pragma unroll
        for (int u = 0; u < U_; ++u) { float v = p[u]; float v2 = v * v; s4 += v2 * v2; }
        sum += expf(2.0f * sqrtf(sqrtf(s4)) - mx);
    }
    const float inv = 1.0f / sum;
    for (int s = 0; s < S_; ++s) {
        float o = 0.f;
        if (mb[s]) {
            const float* p = smb + s * (NH_ * U_);
            float s4 = 0.f;
#pragma unroll
            for (int u = 0; u < U_; ++u) { float v = p[u]; float v2 = v * v; s4 += v2 * v2; }
            o = expf(2.0f * sqrtf(sqrtf(s4)) - mx) * inv;
        }
        ob[s * NH_] = o;
    }
}

// ---------------------------------------------------------------------------
// a_pre[b,s,h] = am2[b,s,h/64] * V[b,s,h] * mask[b,s]
// ---------------------------------------------------------------------------
__global__ void apre_kernel(const float* __restrict__ am2,
                            const float* __restrict__ V,
                            const int* __restrict__ mask,
                            float* __restrict__ Ap)
{
    const size_t idx = (size_t)blockIdx.x * blockDim.x + threadIdx.x;
    if (idx >= (size_t)B_ * S_ * H_) return;
    const int b = (int)(idx / (S_ * H_));
    const int rem = (int)(idx % (S_ * H_));
    const int s = rem >> 8;
    const int h = rem & (H_ - 1);
    const float mval = mask[(size_t)b * S_ + s] ? 1.0f : 0.0f;
    Ap[idx] = am2[(size_t)b * (S_ * NH_) + s * NH_ + (h >> 6)] * V[idx] * mval;
}

// ---------------------------------------------------------------------------
// out = LayerNorm(D + input) * g + b, per row of 256, eps = 1e-12
// ---------------------------------------------------------------------------
__global__ void __launch_bounds__(256) add_ln_kernel(
    const float* __restrict__ D, const float* __restrict__ X0,
    const float* __restrict__ g, const float* __restrict__ bb,
    float* __restrict__ out)
{
    __shared__ float red[256];
    const size_t row = blockIdx.x;
    const int h = threadIdx.x;
    const float x = D[row * H_ + h] + X0[row * H_ + h];
    red[h] = x;
    __syncthreads();
    for (int off = 128; off > 0; off >>= 1) {
        if (h < off) red[h] += red[h + off];
        __syncthreads();
    }
    const float mu = red[0] * (1.0f / H_);
    __syncthreads();
    const float dv = x - mu;
    red[h] = dv * dv;
    __syncthreads();
    for (int off = 128; off > 0; off >>= 1) {
        if (h < off) red[h] += red[h + off];
        __syncthreads();
    }
    const float var = red[0] * (1.0f / H_);
    out[row * H_ + h] = dv / sqrtf(var + 1e-12f) * g[h] + bb[h];
}

// ---------------------------------------------------------------------------
extern "C" void kernel_launch(void* const* d_in, const int* in_sizes, int n_in,
                              void* d_out, int out_size, void* d_ws, size_t ws_size,
                              hipStream_t stream)
{
    (void)in_sizes; (void)n_in; (void)out_size; (void)ws_size;

    const float* input  = (const float*)d_in[0];
    const int*   scaled = (const int*)d_in[1];
    const int*   mask   = (const int*)d_in[2];
    const float* mq_ws[U_];
    const float* mq_bs[U_];
    for (int i = 0; i < U_; ++i) {
        mq_ws[i] = (const float*)d_in[3 + i];
        mq_bs[i] = (const float*)d_in[8 + i];
    }
    const float* Wq  = (const float*)d_in[13];
    const float* bq  = (const float*)d_in[14];
    const float* Wk  = (const float*)d_in[15];
    const float* bk  = (const float*)d_in[16];
    const float* Wv  = (const float*)d_in[17];
    const float* bv  = (const float*)d_in[18];
    const float* Wd  = (const float*)d_in[19];
    const float* bd  = (const float*)d_in[20];
    const float* lng = (const float*)d_in[21];
    const float* lnb = (const float*)d_in[22];

    float* out    = (float*)d_out;
    float* attmap = out + (size_t)B_ * S_ * H_;   // (B,NH,S,U) flat == softmax buf

    // workspace layout (floats)
    float* ws  = (float*)d_ws;
    float* cat = ws;                                   // B * 1280
    float* mq  = cat + (size_t)B_ * U_ * H_;           // B * 1280
    float* Qb  = mq  + (size_t)B_ * U_ * H_;           // B * 1280
    float* Kb  = Qb  + (size_t)B_ * U_ * H_;           // B*S*H (reused as a_pre)
    float* Vb  = Kb  + (size_t)B_ * S_ * H_;           // B*S*H (reused as D)
    float* Lb  = Vb  + (size_t)B_ * S_ * H_;           // B*S*20
    float* am2 = Lb  + (size_t)B_ * S_ * NH_ * U_;     // B*S*NH

    // 1) gather history rows
    gather_kernel<<<B_, H_, 0, stream>>>(input, scaled, cat);

    // 2) mq projections: 5 GEMMs (1024 x (i+1)*256) x ((i+1)*256 x 256)
    for (int i = 0; i < U_; ++i) {
        gemm_wmma_f32<<<dim3(B_ / 64, 2), 128, 0, stream>>>(
            cat, U_ * H_, mq_ws[i], mq_bs[i], mq + i * H_, U_ * H_, (i + 1) * H_);
    }

    // 3) normalize mq over U
    mqnorm_kernel<<<B_, H_, 0, stream>>>(mq);

    // 4) Q = mq @ Wq.T + bq   (5120 x 256 x 256)
    gemm_wmma_f32<<<dim3((B_ * U_) / 64, 2), 128, 0, stream>>>(
        mq, H_, Wq, bq, Qb, H_, H_);

    // 5) K, V big GEMMs (204800 x 256 x 256)
    gemm_wmma_f32<<<dim3((B_ * S_) / 64, 2), 128, 0, stream>>>(
        input, H_, Wk, bk, Kb, H_, H_);
    gemm_wmma_f32<<<dim3((B_ * S_) / 64, 2), 128, 0, stream>>>(
        input, H_, Wv, bv, Vb, H_, H_);

    // 6) sigmoid attention logits
    logits_kernel<<<B_, 256, 0, stream>>>(Qb, Kb, Lb);

    // 7) softmax over S -> att_map output (also consumed by pooling)
    softmax_seq_kernel<<<(B_ * NH_ * U_ + 255) / 256, 256, 0, stream>>>(Lb, attmap);

    // 8) L4 pooling + masked softmax over S
    pooled_softmax_kernel<<<(B_ * NH_ + 255) / 256, 256, 0, stream>>>(attmap, mask, am2);

    // 9) a_pre = am2 * V * mask   (write into Kb, K no longer needed)
    apre_kernel<<<(B_ * S_ * H_ + 255) / 256, 256, 0, stream>>>(am2, Vb, mask, Kb);

    // 10) D = a_pre @ Wd.T + bd   (write into Vb, V no longer needed)
    gemm_wmma_f32<<<dim3((B_ * S_) / 64, 2), 128, 0, stream>>>(
        Kb, H_, Wd, bd, Vb, H_, H_);

    // 11) residual + LayerNorm
    add_ln_kernel<<<B_ * S_, H_, 0, stream>>>(Vb, input, lng, lnb, out);
}